// LangGuidedRelationEncoder_80290118632225
// MI455X (gfx1250) — compile-verified
//
#include <hip/hip_runtime.h>
#include <math.h>

// Problem constants (match reference)
#define Bsz 8
#define Nn  256
#define Cc  128
#define Tt  64
#define Hh  4
#define Gg  12
#define Dd  32
#define EPSF 1e-6f

typedef __attribute__((ext_vector_type(2))) float v2f;
typedef __attribute__((ext_vector_type(8))) float v8f;

// D = A(16x4,f32) * B(4x16,f32) + C(16x16,f32)   -> v_wmma_f32_16x16x4_f32
__device__ __forceinline__ v8f wmma4(v2f a, v2f b, v8f c) {
  return __builtin_amdgcn_wmma_f32_16x16x4_f32(false, a, false, b, (short)0, c,
                                               false, false);
}

// ---------------------------------------------------------------------------
// Generic GEMM: out[M x Nc] = A[M x K] @ W[K x Nc] + bias[Nc]
// One wave per 16x16 output tile. K must be a multiple of 4.
// grid = (M/16, Nc/16), block = 32
// ---------------------------------------------------------------------------
__global__ void __launch_bounds__(32)
gemm_bias_kernel(const float* __restrict__ A, const float* __restrict__ W,
                 const float* __restrict__ bias, float* __restrict__ out,
                 int M, int K, int Nc) {
  const int lane = threadIdx.x & 31;
  const int half = lane >> 4;   // which K pair / which row half for C
  const int ln   = lane & 15;   // M index for A, N index for B/C
  const int row0 = blockIdx.x * 16;
  const int col0 = blockIdx.y * 16;

  v8f acc;
  const float bv = bias[col0 + ln];
#pragma unroll
  for (int r = 0; r < 8; ++r) acc[r] = bv;

  for (int k0 = 0; k0 < K; k0 += 4) {
    v2f a, b;
    const float* Ap = A + (size_t)(row0 + ln) * K + (k0 + 2 * half);
    a[0] = Ap[0];
    a[1] = Ap[1];
    const float* Wp = W + (size_t)(k0 + 2 * half) * Nc + (col0 + ln);
    b[0] = Wp[0];
    b[1] = Wp[Nc];
    acc = wmma4(a, b, acc);
  }

#pragma unroll
  for (int r = 0; r < 8; ++r) {
    const int row = r + 8 * half;
    out[(size_t)(row0 + row) * Nc + col0 + ln] = acc[r];
  }
}

// ---------------------------------------------------------------------------
// Fused: geometry -> rel_feat -> lang cross-attn -> bias head
// grid = (N, B), block = 128 (4 waves). Each wave handles a 16-wide j tile,
// iterating 4 times to cover all 256 j for this (b, i).
// ---------------------------------------------------------------------------
__global__ void __launch_bounds__(128)
relbias_kernel(const float* __restrict__ centers,
               const float* __restrict__ corners,
               const float* __restrict__ Wrel, const float* __restrict__ brel,
               const float* __restrict__ ktxt, const float* __restrict__ vtxt,
               const float* __restrict__ Wbias, const float* __restrict__ bbias,
               float* __restrict__ biasout) {
  __shared__ float ldsG[4][16][Gg];   // geometry features (16 x 12)
  __shared__ float ldsR[4][16][Cc];   // rel_feat, later rel_feat + ctx
  __shared__ float ldsA[4][16][Tt];   // scores / alpha

  const int i    = blockIdx.x;
  const int b    = blockIdx.y;
  const int w    = threadIdx.x >> 5;
  const int lane = threadIdx.x & 31;
  const int half = lane >> 4;
  const int ln   = lane & 15;

  // sizes for node i (same for all lanes; cheap)
  float sxi, syi, szi;
  {
    const float* cp = corners + (size_t)(b * Nn + i) * 24;
    float mx0 = -1e30f, mx1 = -1e30f, mx2 = -1e30f;
    float mn0 = 1e30f, mn1 = 1e30f, mn2 = 1e30f;
#pragma unroll
    for (int c = 0; c < 8; ++c) {
      float x = cp[c * 3 + 0], y = cp[c * 3 + 1], z = cp[c * 3 + 2];
      mx0 = fmaxf(mx0, x); mn0 = fminf(mn0, x);
      mx1 = fmaxf(mx1, y); mn1 = fminf(mn1, y);
      mx2 = fmaxf(mx2, z); mn2 = fminf(mn2, z);
    }
    sxi = mx0 - mn0; syi = mx1 - mn1; szi = mx2 - mn2;
  }
  const float cxi = centers[(size_t)(b * Nn + i) * 3 + 0];
  const float cyi = centers[(size_t)(b * Nn + i) * 3 + 1];
  const float czi = centers[(size_t)(b * Nn + i) * 3 + 2];

  const float rs128 = 0.08838834764831845f;  // 1/sqrt(128)

  for (int jo = 0; jo < 4; ++jo) {
    const int j0 = jo * 64 + w * 16;

    // ---- stage 1: geometry features for 16 j rows (lanes 0..15) ----
    if (lane < 16) {
      const int j = j0 + lane;
      const float cxj = centers[(size_t)(b * Nn + j) * 3 + 0];
      const float cyj = centers[(size_t)(b * Nn + j) * 3 + 1];
      const float czj = centers[(size_t)(b * Nn + j) * 3 + 2];
      const float r0 = cxj - cxi, r1 = cyj - cyi, r2 = czj - czi;
      const float dist  = sqrtf(r0 * r0 + r1 * r1 + r2 * r2);
      const float horiz = sqrtf(r0 * r0 + r1 * r1);
      const float cose  = r2 / (dist + EPSF);
      const float dlog  = logf(dist + EPSF);
      // sizes for node j
      const float* cp = corners + (size_t)(b * Nn + j) * 24;
      float mx0 = -1e30f, mx1 = -1e30f, mx2 = -1e30f;
      float mn0 = 1e30f, mn1 = 1e30f, mn2 = 1e30f;
#pragma unroll
      for (int c = 0; c < 8; ++c) {
        float x = cp[c * 3 + 0], y = cp[c * 3 + 1], z = cp[c * 3 + 2];
        mx0 = fmaxf(mx0, x); mn0 = fminf(mn0, x);
        mx1 = fmaxf(mx1, y); mn1 = fminf(mn1, y);
        mx2 = fmaxf(mx2, z); mn2 = fminf(mn2, z);
      }
      const float sxj = mx0 - mn0, syj = mx1 - mn1, szj = mx2 - mn2;
      float* g = &ldsG[w][lane][0];
      g[0] = r0; g[1] = r1; g[2] = r2;
      g[3] = dlog; g[4] = horiz; g[5] = cose;
      g[6] = sxj - sxi; g[7] = syj - syi; g[8] = szj - szi;
      g[9]  = sxj / (sxi + EPSF);
      g[10] = syj / (syi + EPSF);
      g[11] = szj / (szi + EPSF);
    }
    __syncthreads();

    // ---- stage 2: rel_feat = geom(16x12) @ Wrel(12x128) + brel ----
    for (int ct = 0; ct < 8; ++ct) {
      const int c0 = ct * 16;
      v8f acc;
      const float bv = brel[c0 + ln];
#pragma unroll
      for (int r = 0; r < 8; ++r) acc[r] = bv;
#pragma unroll
      for (int k0 = 0; k0 < Gg; k0 += 4) {
        v2f a, bb;
        a[0] = ldsG[w][ln][k0 + 2 * half];
        a[1] = ldsG[w][ln][k0 + 2 * half + 1];
        bb[0] = Wrel[(size_t)(k0 + 2 * half) * Cc + c0 + ln];
        bb[1] = Wrel[(size_t)(k0 + 2 * half + 1) * Cc + c0 + ln];
        acc = wmma4(a, bb, acc);
      }
#pragma unroll
      for (int r = 0; r < 8; ++r) ldsR[w][r + 8 * half][c0 + ln] = acc[r];
    }
    __syncthreads();

    // ---- stage 3: scores = rel_feat(16x128) @ ktxt^T(128x64) * rs128 ----
    for (int tt = 0; tt < 4; ++tt) {
      const int t0 = tt * 16;
      v8f acc;
#pragma unroll
      for (int r = 0; r < 8; ++r) acc[r] = 0.f;
      for (int k0 = 0; k0 < Cc; k0 += 4) {
        v2f a, bb;
        a[0] = ldsR[w][ln][k0 + 2 * half];
        a[1] = ldsR[w][ln][k0 + 2 * half + 1];
        const float* kp = ktxt + (size_t)(b * Tt + t0 + ln) * Cc + k0 + 2 * half;
        bb[0] = kp[0];
        bb[1] = kp[1];
        acc = wmma4(a, bb, acc);
      }
#pragma unroll
      for (int r = 0; r < 8; ++r)
        ldsA[w][r + 8 * half][t0 + ln] = acc[r] * rs128;
    }
    __syncthreads();

    // ---- stage 4: softmax over T=64, 2 lanes per row ----
    {
      const int row = ln, seg = half;
      float mx = -1e30f;
#pragma unroll
      for (int t = 0; t < 32; ++t) mx = fmaxf(mx, ldsA[w][row][seg * 32 + t]);
      mx = fmaxf(mx, __shfl_xor(mx, 16, 32));
      float sum = 0.f;
#pragma unroll
      for (int t = 0; t < 32; ++t) {
        float e = __expf(ldsA[w][row][seg * 32 + t] - mx);
        ldsA[w][row][seg * 32 + t] = e;
        sum += e;
      }
      sum += __shfl_xor(sum, 16, 32);
      const float inv = 1.f / sum;
#pragma unroll
      for (int t = 0; t < 32; ++t) ldsA[w][row][seg * 32 + t] *= inv;
    }
    __syncthreads();

    // ---- stage 5: ctx = alpha(16x64) @ vtxt(64x128); rel += ctx ----
    for (int ct = 0; ct < 8; ++ct) {
      const int c0 = ct * 16;
      v8f acc;
#pragma unroll
      for (int r = 0; r < 8; ++r) acc[r] = 0.f;
      for (int k0 = 0; k0 < Tt; k0 += 4) {
        v2f a, bb;
        a[0] = ldsA[w][ln][k0 + 2 * half];
        a[1] = ldsA[w][ln][k0 + 2 * half + 1];
        const float* vp = vtxt + (size_t)(b * Tt + k0 + 2 * half) * Cc + c0 + ln;
        bb[0] = vp[0];
        bb[1] = vp[Cc];
        acc = wmma4(a, bb, acc);
      }
#pragma unroll
      for (int r = 0; r < 8; ++r) ldsR[w][r + 8 * half][c0 + ln] += acc[r];
    }
    __syncthreads();

    // ---- stage 6: bias = (rel+ctx)(16x128) @ Wbias(128x4) + bbias ----
    {
      v8f acc;
      const float bv = (ln < Hh) ? bbias[ln] : 0.f;
#pragma unroll
      for (int r = 0; r < 8; ++r) acc[r] = bv;
      for (int k0 = 0; k0 < Cc; k0 += 4) {
        v2f a, bb;
        a[0] = ldsR[w][ln][k0 + 2 * half];
        a[1] = ldsR[w][ln][k0 + 2 * half + 1];
        bb[0] = (ln < Hh) ? Wbias[(size_t)(k0 + 2 * half) * Hh + ln] : 0.f;
        bb[1] = (ln < Hh) ? Wbias[(size_t)(k0 + 2 * half + 1) * Hh + ln] : 0.f;
        acc = wmma4(a, bb, acc);
      }
      if (ln < Hh) {
#pragma unroll
        for (int r = 0; r < 8; ++r) {
          const int row = r + 8 * half;
          const int j = j0 + row;
          biasout[((size_t)(b * Hh + ln) * Nn + i) * Nn + j] = acc[r];
        }
      }
    }
    __syncthreads();
  }
}

// ---------------------------------------------------------------------------
// Self-attention: per (b, h, 16-row q tile). One wave per block.
// grid = (N/16, H, B), block = 32
// ---------------------------------------------------------------------------
__global__ void __launch_bounds__(32)
attn_kernel(const float* __restrict__ q, const float* __restrict__ k,
            const float* __restrict__ v, const float* __restrict__ bias,
            float* __restrict__ outp) {
  __shared__ float ldsL[16][Nn];  // logits -> attn weights (16 x 256)

  const int i0   = blockIdx.x * 16;
  const int h    = blockIdx.y;
  const int b    = blockIdx.z;
  const int lane = threadIdx.x & 31;
  const int half = lane >> 4;
  const int ln   = lane & 15;
  const float scale = 0.1767766952966369f;  // 1/sqrt(32)

  // logits = q(16x32) @ k^T(32x256) * scale + bias
  for (int nt = 0; nt < 16; ++nt) {
    const int n0 = nt * 16;
    v8f acc;
#pragma unroll
    for (int r = 0; r < 8; ++r) acc[r] = 0.f;
#pragma unroll
    for (int k0 = 0; k0 < Dd; k0 += 4) {
      v2f a, bb;
      const float* qp = q + (size_t)(b * Nn + i0 + ln) * Cc + h * Dd + k0 + 2 * half;
      a[0] = qp[0];
      a[1] = qp[1];
      const float* kp = k + (size_t)(b * Nn + n0 + ln) * Cc + h * Dd + k0 + 2 * half;
      bb[0] = kp[0];
      bb[1] = kp[1];
      acc = wmma4(a, bb, acc);
    }
#pragma unroll
    for (int r = 0; r < 8; ++r) {
      const int row = r + 8 * half;
      ldsL[row][n0 + ln] =
          acc[r] * scale +
          bias[((size_t)(b * Hh + h) * Nn + (i0 + row)) * Nn + n0 + ln];
    }
  }
  __syncthreads();

  // softmax over 256: 2 lanes per row, 128 elems each
  {
    const int row = ln, seg = half;
    float mx = -1e30f;
    for (int t = 0; t < 128; ++t) mx = fmaxf(mx, ldsL[row][seg * 128 + t]);
    mx = fmaxf(mx, __shfl_xor(mx, 16, 32));
    float sum = 0.f;
    for (int t = 0; t < 128; ++t) {
      float e = __expf(ldsL[row][seg * 128 + t] - mx);
      ldsL[row][seg * 128 + t] = e;
      sum += e;
    }
    sum += __shfl_xor(sum, 16, 32);
    const float inv = 1.f / sum;
    for (int t = 0; t < 128; ++t) ldsL[row][seg * 128 + t] *= inv;
  }
  __syncthreads();

  // out = attn(16x256) @ v(256x32)
  for (int ct = 0; ct < 2; ++ct) {
    const int c0 = ct * 16;
    v8f acc;
#pragma unroll
    for (int r = 0; r < 8; ++r) acc[r] = 0.f;
    for (int k0 = 0; k0 < Nn; k0 += 4) {
      v2f a, bb;
      a[0] = ldsL[ln][k0 + 2 * half];
      a[1] = ldsL[ln][k0 + 2 * half + 1];
      const float* vp = v + (size_t)(b * Nn + k0 + 2 * half) * Cc + h * Dd + c0 + ln;
      bb[0] = vp[0];
      bb[1] = vp[Cc];
      acc = wmma4(a, bb, acc);
    }
#pragma unroll
    for (int r = 0; r < 8; ++r) {
      const int row = r + 8 * half;
      outp[(size_t)(b * Nn + i0 + row) * Cc + h * Dd + c0 + ln] = acc[r];
    }
  }
}

// ---------------------------------------------------------------------------
extern "C" void kernel_launch(void* const* d_in, const int* in_sizes, int n_in,
                              void* d_out, int out_size, void* d_ws,
                              size_t ws_size, hipStream_t stream) {
  (void)in_sizes; (void)n_in; (void)out_size; (void)ws_size;
  const float* feat    = (const float*)d_in[0];
  const float* centers = (const float*)d_in[1];
  const float* corners = (const float*)d_in[2];
  const float* lang    = (const float*)d_in[3];
  const float* Wq = (const float*)d_in[4];  const float* bq = (const float*)d_in[5];
  const float* Wk = (const float*)d_in[6];  const float* bk = (const float*)d_in[7];
  const float* Wv = (const float*)d_in[8];  const float* bv = (const float*)d_in[9];
  const float* Wo = (const float*)d_in[10]; const float* bo = (const float*)d_in[11];
  const float* Wrel = (const float*)d_in[12]; const float* brel = (const float*)d_in[13];
  const float* Wtk  = (const float*)d_in[14]; const float* btk  = (const float*)d_in[15];
  const float* Wtv  = (const float*)d_in[16]; const float* btv  = (const float*)d_in[17];
  const float* Wbias = (const float*)d_in[18]; const float* bbias = (const float*)d_in[19];

  float* ws = (float*)d_ws;
  const size_t MR = (size_t)Bsz * Nn;  // 2048 rows
  float* qb      = ws;
  float* kb      = qb + MR * Cc;
  float* vb      = kb + MR * Cc;
  float* ktxt    = vb + MR * Cc;
  float* vtxt    = ktxt + (size_t)Bsz * Tt * Cc;
  float* biasb   = vtxt + (size_t)Bsz * Tt * Cc;
  float* attnout = biasb + (size_t)Bsz * Hh * Nn * Nn;

  // q / k / v projections  (2048x128 @ 128x128)
  dim3 gqkv(MR / 16, Cc / 16);
  gemm_bias_kernel<<<gqkv, 32, 0, stream>>>(feat, Wq, bq, qb, MR, Cc, Cc);
  gemm_bias_kernel<<<gqkv, 32, 0, stream>>>(feat, Wk, bk, kb, MR, Cc, Cc);
  gemm_bias_kernel<<<gqkv, 32, 0, stream>>>(feat, Wv, bv, vb, MR, Cc, Cc);

  // text projections (512x128 @ 128x128)
  dim3 gtxt((size_t)Bsz * Tt / 16, Cc / 16);
  gemm_bias_kernel<<<gtxt, 32, 0, stream>>>(lang, Wtk, btk, ktxt, Bsz * Tt, Cc, Cc);
  gemm_bias_kernel<<<gtxt, 32, 0, stream>>>(lang, Wtv, btv, vtxt, Bsz * Tt, Cc, Cc);

  // fused geometry + language cross-attention -> bias(B,H,N,N)
  relbias_kernel<<<dim3(Nn, Bsz), 128, 0, stream>>>(
      centers, corners, Wrel, brel, ktxt, vtxt, Wbias, bbias, biasb);

  // biased multi-head self-attention
  attn_kernel<<<dim3(Nn / 16, Hh, Bsz), 32, 0, stream>>>(qb, kb, vb, biasb,
                                                         attnout);

  // final projection -> d_out
  gemm_bias_kernel<<<gqkv, 32, 0, stream>>>(attnout, Wo, bo, (float*)d_out, MR,
                                            Cc, Cc);
}